// BaseAtt_15968688406991
// MI455X (gfx1250) — compile-verified
//
#include <hip/hip_runtime.h>

// Problem sizes (fixed by the reference)
#define B_  4096
#define K_  200
#define D_  128
#define FD_ 1024
#define MASK_FILL (-4294967295.0f)   // -(2^32)+1

typedef __attribute__((ext_vector_type(2))) float v2f;
typedef __attribute__((ext_vector_type(8))) float v8f;
typedef __attribute__((ext_vector_type(4))) int   v4i;

#define GAS __attribute__((address_space(1)))   // global
#define LAS __attribute__((address_space(3)))   // LDS

// -----------------------------------------------------------------------------
// Kernel 1: target[B,D] = target_feats[B,FD] @ W[D,FD]^T
// via V_WMMA_F32_16X16X4_F32 (fp32 WMMA, K-step 4).
// Block = 256 threads (8 wave32). Each block computes a 16-row M tile;
// wave w computes the 16-col N tile at n0 = 16*w. K loop over FD=1024.
//
// A layout (16x4 f32): lanes 0-15 -> M=0..15 holding K=0 (v0) / K=1 (v1);
//                      lanes 16-31 -> same M rows, K=2 (v0) / K=3 (v1).
// B layout (4x16 f32): lanes 0-15 -> N=0..15 of K-row 0 (v0) / 1 (v1);
//                      lanes 16-31 -> K-row 2 (v0) / 3 (v1).
// C layout (16x16 f32, 8 VGPRs): v[r]: lanes 0-15 = M=r, lanes 16-31 = M=r+8.
// -----------------------------------------------------------------------------
__global__ __launch_bounds__(256) void target_gemm_wmma(
    const float* __restrict__ TF, const float* __restrict__ W,
    float* __restrict__ TGT) {
  const int tid  = threadIdx.x;
  const int wave = tid >> 5;
  const int lane = tid & 31;
  const int m0 = blockIdx.x * 16;
  const int n0 = wave * 16;
  const int rc = lane & 15;   // A row (lanes' M) / B col (lanes' N)
  const int hi = lane >> 4;   // lane half selects K offset {0,2}

  const v2f* A2 = (const v2f*)TF;
  const v2f* B2 = (const v2f*)W;
  const long aBase = ((long)(m0 + rc) * FD_ + hi * 2) >> 1;  // float2 units
  const long bBase = ((long)(n0 + rc) * FD_ + hi * 2) >> 1;

  v8f c = {};
#pragma unroll 4
  for (int f = 0; f < FD_; f += 4) {
    v2f a = A2[aBase + (f >> 1)];
    v2f b = B2[bBase + (f >> 1)];
    // 8 args: (neg_a, A, neg_b, B, c_mod, C, reuse_a, reuse_b)
    c = __builtin_amdgcn_wmma_f32_16x16x4_f32(false, a, false, b, (short)0, c,
                                              false, false);
  }

  float* outp = TGT + (long)(m0 + hi * 8) * D_ + n0 + rc;
#pragma unroll
  for (int r = 0; r < 8; ++r) outp[r * D_] = c[r];
}

// -----------------------------------------------------------------------------
// Kernel 2: fused  alpha = NF[b] @ target[b]  -> mask -> softmax ->
//           neighbor_embd = alpha @ NF[b], label_embd = alpha @ NL[b]
// One 256-thread workgroup per batch row. NF[b] (200x128 f32 = 100 KB) is
// staged once into LDS (padded stride 132 floats to dodge bank conflicts)
// via GLOBAL_LOAD_ASYNC_TO_LDS_B128 (ASYNCcnt-tracked DMA, no VGPR round
// trip); NL[b] is streamed coalesced from global exactly once.
// -----------------------------------------------------------------------------
#define S_ 132  // padded LDS row stride in floats (132*4 bytes, 16B aligned)

__global__ __launch_bounds__(256) void fused_attention(
    const float* __restrict__ NF, const float* __restrict__ NL,
    const int* __restrict__ MASK, const float* __restrict__ TGT,
    float* __restrict__ OUT) {
  extern __shared__ float lds_nf[];  // K_ * S_ floats = 105,600 B
  __shared__ float s_tgt[D_];
  __shared__ float s_alpha[256];
  __shared__ float s_red[8];
  __shared__ float s_accN[256];
  __shared__ float s_accL[256];

  const int tid  = threadIdx.x;
  const int wave = tid >> 5;
  const int lane = tid & 31;
  const long b = blockIdx.x;

  // ---- Phase 0: stage NF[b] tile into LDS (async DMA, fully coalesced) ----
  {
    float4* dst = (float4*)lds_nf;
    const float4* src = (const float4*)(NF + b * (long)(K_ * D_));
    const int q = tid & 31;  // float4 index within a 128-float row
#if __has_builtin(__builtin_amdgcn_global_load_async_to_lds_b128)
    for (int r = tid >> 5; r < K_; r += 8)
      __builtin_amdgcn_global_load_async_to_lds_b128(
          (GAS v4i*)(src + r * (D_ / 4) + q),
          (LAS v4i*)(dst + r * (S_ / 4) + q), 0, 0);
#else
    for (int r = tid >> 5; r < K_; r += 8)
      dst[r * (S_ / 4) + q] = src[r * (D_ / 4) + q];
#endif
    // overlap: pull target row while the async engine fills LDS
    if (tid < D_) s_tgt[tid] = TGT[b * D_ + tid];
#if __has_builtin(__builtin_amdgcn_global_load_async_to_lds_b128)
#if __has_builtin(__builtin_amdgcn_s_wait_asynccnt)
    __builtin_amdgcn_s_wait_asynccnt(0);
#else
    asm volatile("s_wait_asynccnt 0x0" ::: "memory");
#endif
#endif
  }
  __syncthreads();

  // ---- Phase 1: alpha[k] = <NF[b,k,:], target[b,:]>, then mask ----
  {
    const float t0 = s_tgt[lane], t1 = s_tgt[lane + 32],
                t2 = s_tgt[lane + 64], t3 = s_tgt[lane + 96];
    for (int k = wave; k < K_; k += 8) {
      const float* rowp = lds_nf + k * S_;
      float p = rowp[lane] * t0 + rowp[lane + 32] * t1 +
                rowp[lane + 64] * t2 + rowp[lane + 96] * t3;
#pragma unroll
      for (int off = 16; off > 0; off >>= 1) p += __shfl_xor(p, off);
      if (lane == 0)
        s_alpha[k] = (MASK[b * K_ + k] > 0) ? p : MASK_FILL;
    }
  }
  __syncthreads();

  // ---- Phase 2: numerically-stable softmax over K=200 ----
  {
    const float v = (tid < K_) ? s_alpha[tid] : MASK_FILL;
    float m = v;
#pragma unroll
    for (int off = 16; off > 0; off >>= 1) m = fmaxf(m, __shfl_xor(m, off));
    if (lane == 0) s_red[wave] = m;
    __syncthreads();
    if (tid == 0) {
      float mm = s_red[0];
#pragma unroll
      for (int i = 1; i < 8; ++i) mm = fmaxf(mm, s_red[i]);
      s_red[0] = mm;
    }
    __syncthreads();
    const float rowmax = s_red[0];
    __syncthreads();

    const float e = (tid < K_) ? __expf(v - rowmax) : 0.0f;
    float s = e;
#pragma unroll
    for (int off = 16; off > 0; off >>= 1) s += __shfl_xor(s, off);
    if (lane == 0) s_red[wave] = s;
    __syncthreads();
    if (tid == 0) {
      float ss = s_red[0];
#pragma unroll
      for (int i = 1; i < 8; ++i) ss += s_red[i];
      s_red[0] = ss;
    }
    __syncthreads();
    const float inv = 1.0f / s_red[0];
    if (tid < K_) s_alpha[tid] = e * inv;
  }
  __syncthreads();

  // ---- Phase 3: weighted sums (NF from LDS, NL streamed from global) ----
  {
    const int d = tid & (D_ - 1);  // 0..127
    const int h = tid >> 7;        // k parity (0/1)
    float accN = 0.0f, accL = 0.0f;
    for (int k = h; k < K_; k += 2) {
      const float a = s_alpha[k];
      accN += a * lds_nf[k * S_ + d];
      accL += a * NL[(b * K_ + k) * (long)D_ + d];
    }
    s_accN[tid] = accN;
    s_accL[tid] = accL;
    __syncthreads();
    if (tid < D_) {
      OUT[b * D_ + tid] = s_accN[tid] + s_accN[tid + 128];
      OUT[(long)B_ * D_ + b * D_ + tid] = s_accL[tid] + s_accL[tid + 128];
    }
  }
}

// -----------------------------------------------------------------------------
extern "C" void kernel_launch(void* const* d_in, const int* in_sizes, int n_in,
                              void* d_out, int out_size, void* d_ws,
                              size_t ws_size, hipStream_t stream) {
  const float* tf   = (const float*)d_in[0];  // [B, FD]
  const float* nf   = (const float*)d_in[1];  // [B, K, D]
  const float* nl   = (const float*)d_in[2];  // [B, K, D]
  const int*   mask = (const int*)d_in[3];    // [B, K]
  const float* W    = (const float*)d_in[4];  // [D, FD]
  float* out = (float*)d_out;                 // [B,D] ++ [B,D]
  float* tgt = (float*)d_ws;                  // scratch: target [B, D] (2 MB)

  target_gemm_wmma<<<B_ / 16, 256, 0, stream>>>(tf, W, tgt);

  const size_t dynLds = (size_t)K_ * S_ * sizeof(float);  // 105,600 B
  fused_attention<<<B_, 256, dynLds, stream>>>(nf, nl, mask, tgt, out);
}